// Kuramoto_Solver_78812649881984
// MI455X (gfx1250) — compile-verified
//
#include <hip/hip_runtime.h>
#include <math.h>

typedef __attribute__((ext_vector_type(2))) float v2f;
typedef __attribute__((ext_vector_type(4))) float v4f;
typedef __attribute__((ext_vector_type(8))) float v8f;
typedef __attribute__((ext_vector_type(4))) unsigned int v4u;
typedef __attribute__((ext_vector_type(8))) unsigned int v8u;

#define Bdim   64
#define Mdim   116
#define Cdim   512
#define MC     (Mdim*Cdim)   /* 59392 */
#define HISTN  24
#define BETA_C 0.25f

// fp32 WMMA: D(16x16,f32) = A(16x4,f32) x B(4x16,f32) + C
static __device__ __forceinline__ v8f wmma4(v2f a, v2f b, v8f c) {
  return __builtin_amdgcn_wmma_f32_16x16x4_f32(false, a, false, b, (short)0, c,
                                               false, false);
}

// ---------------------------------------------------------------------------
// Tensor Data Mover: 2D tile (f32) global -> LDS. Issue once per wave
// (EXEC is ignored by TDM). D# built per CDNA5 ISA:
//   group0: [1:0]=count=1, [63:32]=lds_addr, [120:64]=global_addr, [127:126]=2
//   group1: data_size=2(4B), pad ctl, tensor dims (huge; tile is always
//           in-bounds), tile_dim0/1, tensor_dim0_stride.
// ---------------------------------------------------------------------------
static __device__ __forceinline__ void tdm_load_2d(
    const void* gsrc, unsigned lds_byte_off, unsigned tile_d0, unsigned tile_d1,
    unsigned long long stride0_elems, unsigned pad_en, unsigned pad_code) {
  unsigned long long ga = (unsigned long long)gsrc;
  v4u g0;
  g0.x = 1u;                                          // count=1, user-mode
  g0.y = lds_byte_off;                                // lds_addr
  g0.z = (unsigned)ga;                                // global_addr[31:0]
  g0.w = (unsigned)((ga >> 32) & 0x01ffffffull) | (2u << 30);  // [56:32]|type=2
  v8u g1;
  g1.s0 = (2u << 16) | (pad_en << 20) | (pad_code << 22);  // 4B elems, pad ctl
  g1.s1 = 0xffff0000u;                 // tensor_dim0 = 0x7fffffff (lo16)
  g1.s2 = 0xffff7fffu;                 // tensor_dim0 hi16 | tensor_dim1 lo16
  g1.s3 = 0x00007fffu | (tile_d0 << 16);  // tensor_dim1 hi16 | tile_dim0
  g1.s4 = tile_d1 & 0xffffu;              // tile_dim1 (tile_dim2 = 0)
  g1.s5 = (unsigned)stride0_elems;        // tensor_dim0_stride[31:0]
  g1.s6 = (unsigned)(stride0_elems >> 32) & 0xffffu;  // stride0[47:32]
  g1.s7 = 0u;
  asm volatile("tensor_load_to_lds %0, %1" :: "s"(g0), "s"(g1) : "memory");
}

// ---------------------------------------------------------------------------
// P[n][m] = 0.5*(sync[n,m]+sync[m,n])*adj[n,m], zero-padded to 128 rows
// ---------------------------------------------------------------------------
__global__ void prep_p_k(const float* __restrict__ syncp,
                         const float* __restrict__ adj,
                         float* __restrict__ P) {
  int i = blockIdx.x * 256 + threadIdx.x;
  if (i >= 128 * Mdim) return;
  int n = i / Mdim, m = i % Mdim;
  P[i] = (n < Mdim)
             ? 0.5f * (syncp[n * Mdim + m] + syncp[m * Mdim + n]) * adj[n * Mdim + m]
             : 0.f;
}

// ---------------------------------------------------------------------------
// GroupNorm over (4 channels x 116 M) per (b, group); writes transposed yt[b,m,c]
// ---------------------------------------------------------------------------
__global__ void groupnorm_k(const float* __restrict__ ysrc,
                            const float* __restrict__ gamma,
                            const float* __restrict__ beta,
                            float* __restrict__ yt) {
  int b = blockIdx.x >> 7;
  int g = blockIdx.x & 127;
  int lane = threadIdx.x;  // block = 32 threads = 1 wave
  const float* yb = ysrc + ((size_t)b * Cdim + g * 4) * Mdim;
  float s = 0.f, s2 = 0.f;
  for (int i = lane; i < 4 * Mdim; i += 32) {
    float v = yb[i];
    s += v; s2 += v * v;
  }
  for (int off = 16; off > 0; off >>= 1) {
    s  += __shfl_xor(s, off, 32);
    s2 += __shfl_xor(s2, off, 32);
  }
  const float inv_n = 1.f / (4.f * Mdim);
  float mu  = s * inv_n;
  float var = s2 * inv_n - mu * mu;
  float rstd = rsqrtf(var + 1e-5f);
  for (int i = lane; i < 4 * Mdim; i += 32) {
    int cc = i / Mdim, m = i % Mdim;
    int c = g * 4 + cc;
    float v = (yb[i] - mu) * rstd * gamma[c] + beta[c];
    yt[((size_t)b * Mdim + m) * Cdim + c] = v;
  }
}

// ---------------------------------------------------------------------------
// One Kuramoto inner step, fused: z=relu(P x)+yt, project, omega, renorm.
// Block = (batch, 128-channel slab), 256 threads = 8 waves.
// P (128x116) and x slab (116x128) staged into LDS by the TDM.
// ---------------------------------------------------------------------------
__global__ __launch_bounds__(256, 1) void step_k(
    const float* __restrict__ P, const float* __restrict__ yt,
    const float* __restrict__ omega_param,
    const float* __restrict__ xsrc, long xsrc_bs,
    float* __restrict__ xdst, long xdst_bs, int prenorm) {
  extern __shared__ float sm[];
  float* Pls  = sm;                   // byte off 0      : 128*116
  float* Xls  = Pls + 128 * Mdim;     // byte off 59392  : [m][c] 116*128
  float* Zls  = Xls + Mdim * 128;     // [m][c] 116*128
  float* dotp = Zls + Mdim * 128;     // 2*128
  float* dotv = dotp + 256;           // 128
  float* omg  = dotv + 128;           // 64

  int tid = threadIdx.x;
  int b = blockIdx.x;
  int cg0 = blockIdx.y * 128;
  const float* xs = xsrc + (size_t)b * xsrc_bs;
  float* xd = xdst + (size_t)b * xdst_bs;

  // wave 0: DMA both operand tiles into LDS via the Tensor Data Mover
  if (tid < 32) {
    tdm_load_2d(P, 0u, (unsigned)Mdim, 128u, (unsigned long long)Mdim, 0u, 0u);
    tdm_load_2d(xs + cg0, (unsigned)(128 * Mdim * 4), 128u, (unsigned)Mdim,
                (unsigned long long)Cdim, 0u, 0u);
    __builtin_amdgcn_s_wait_tensorcnt(0);
  }
  // waves 1-2: omega magnitudes for this block's 64 channel pairs
  if (tid >= 32 && tid < 96) {
    int p = (cg0 >> 1) + (tid - 32);
    float a0 = omega_param[2 * p], a1 = omega_param[2 * p + 1];
    omg[tid - 32] = sqrtf(a0 * a0 + a1 * a1);
  }
  __syncthreads();

  // map_to_sphere in place on first inner step of each L iteration
  if (prenorm) {
    for (int g = tid; g < Mdim * 32; g += 256) {
      int m = g >> 5, q = g & 31;
      v4f v = *(v4f*)(Xls + m * 128 + q * 4);
      float nr = sqrtf(v.x * v.x + v.y * v.y + v.z * v.z + v.w * v.w);
      float inv = 1.f / fmaxf(nr, 1e-12f);
      v.x *= inv; v.y *= inv; v.z *= inv; v.w *= inv;
      *(v4f*)(Xls + m * 128 + q * 4) = v;
    }
    __syncthreads();
  }

  // GEMM: wave w owns n-tile w (rows w*16..w*16+15), 8 c-tiles in registers
  int lane = tid & 31;
  int wv = tid >> 5;
  int lm = lane & 15;
  int kh = (lane >> 4) << 1;  // 0 or 2
  int n0 = wv * 16;
  v8f acc[8] = {};
  for (int k0 = 0; k0 < Mdim; k0 += 4) {
    v2f a;
    a.x = Pls[(n0 + lm) * Mdim + k0 + kh];
    a.y = Pls[(n0 + lm) * Mdim + k0 + kh + 1];
#pragma unroll
    for (int ct = 0; ct < 8; ++ct) {
      v2f bb;
      bb.x = Xls[(k0 + kh) * 128 + ct * 16 + lm];
      bb.y = Xls[(k0 + kh + 1) * 128 + ct * 16 + lm];
      acc[ct] = wmma4(a, bb, acc[ct]);
    }
  }
  // z = relu(wx) + yt  -> LDS
  int rbase = n0 + ((lane >> 4) << 3);
#pragma unroll
  for (int ct = 0; ct < 8; ++ct) {
    int c = ct * 16 + lm;
#pragma unroll
    for (int r = 0; r < 8; ++r) {
      int n = rbase + r;
      if (n < Mdim) {
        float z = acc[ct][r];
        z = z > 0.f ? z : 0.f;
        z += yt[((size_t)b * Mdim + n) * Cdim + cg0 + c];
        Zls[n * 128 + c] = z;
      }
    }
  }
  __syncthreads();

  // projection dot over M per channel (split across 2 halves)
  {
    int c = tid & 127, h = tid >> 7;
    float s = 0.f;
    for (int m = h * 58; m < h * 58 + 58; ++m)
      s += Xls[m * 128 + c] * Zls[m * 128 + c];
    dotp[h * 128 + c] = s;
  }
  __syncthreads();
  if (tid < 128) dotv[tid] = dotp[tid] + dotp[128 + tid];
  __syncthreads();

  // x_new = normalize4(omega(x) + beta*(z - <x,z>_M x)); write = hist slice
  for (int g = tid; g < Mdim * 32; g += 256) {
    int m = g >> 5, q = g & 31;
    int c0 = q * 4;
    float vv[4];
#pragma unroll
    for (int j = 0; j < 4; ++j) {
      int cl = c0 + j;
      float xv = Xls[m * 128 + cl];
      float zv = Zls[m * 128 + cl];
      float phi = zv - dotv[cl] * xv;
      float om = omg[cl >> 1];
      float pa = Xls[m * 128 + (cl ^ 1)];
      float xo = (cl & 1) ? (-om * pa) : (om * pa);
      vv[j] = xo + BETA_C * phi;
    }
    float nr = sqrtf(vv[0] * vv[0] + vv[1] * vv[1] + vv[2] * vv[2] + vv[3] * vv[3]);
    float inv = 1.f / fmaxf(nr, 1e-12f);
    v4f v; v.x = vv[0] * inv; v.y = vv[1] * inv; v.z = vv[2] * inv; v.w = vv[3] * inv;
    *(v4f*)(xd + (size_t)m * Cdim + cg0 + c0) = v;
  }
}

// ---------------------------------------------------------------------------
// f_phi: h = W[2048,512] @ x_b^T, then y[b,cc,m] = ||h[cc*4..cc*4+3, m]|| + pb
// x_b staged row-major by TDM with 1-dword LDS pad per 256 dwords
// (row stride 514 floats) so strided B-operand reads are bank-conflict-free.
// ---------------------------------------------------------------------------
__global__ __launch_bounds__(256, 1) void conv_k(
    const float* __restrict__ xsrc, long xsrc_bs,
    const float* __restrict__ W, const float* __restrict__ cbias,
    const float* __restrict__ pbias, float* __restrict__ yout) {
  extern __shared__ float sm[];
  float* Xrm = sm;  // [m][c] padded: addr(m,c) = m*514 + c + (c>>8)
  int tid = threadIdx.x;
  int b = blockIdx.x;
  const float* xs = xsrc + (size_t)b * xsrc_bs;

  if (tid < 32) {
    // 116 rows x 512 cols, pad_interval code 7 (256 dwords), pad_amount 1 dword
    tdm_load_2d(xs, 0u, (unsigned)Cdim, (unsigned)Mdim,
                (unsigned long long)Cdim, 1u, 7u);
    __builtin_amdgcn_s_wait_tensorcnt(0);
  }
  __syncthreads();

  int lane = tid & 31, wv = tid >> 5;
  int lm = lane & 15, kh = (lane >> 4) << 1;
  int o0 = blockIdx.y * 128 + wv * 16;
  const float* Wr = W + (size_t)(o0 + lm) * Cdim;
  __builtin_prefetch(Wr, 0, 0);
  v8f acc[8] = {};
  for (int k0 = 0; k0 < Cdim; k0 += 4) {
    int kc = k0 + kh;
    v2f a; a.x = Wr[kc]; a.y = Wr[kc + 1];
    const float* xr0 = Xrm + kc + (kc >> 8);
    const float* xr1 = Xrm + (kc + 1) + ((kc + 1) >> 8);
#pragma unroll
    for (int mt = 0; mt < 8; ++mt) {
      int m = mt * 16 + lm;
      v2f bb;
      bb.x = xr0[m * 514];
      bb.y = xr1[m * 514];
      acc[mt] = wmma4(a, bb, acc[mt]);
    }
  }
  // fuse bias + 4-row L2 norm directly from D layout (VGPR r -> rows r / r+8)
  int hi = (lane >> 4) << 3;
  float cb[8];
#pragma unroll
  for (int r = 0; r < 8; ++r) cb[r] = cbias[o0 + hi + r];
  int cc = (o0 + hi) >> 2;
  float pb0 = pbias[cc], pb1 = pbias[cc + 1];
#pragma unroll
  for (int mt = 0; mt < 8; ++mt) {
    int m = mt * 16 + lm;
    if (m < Mdim) {
      float h0 = acc[mt][0] + cb[0], h1 = acc[mt][1] + cb[1];
      float h2 = acc[mt][2] + cb[2], h3 = acc[mt][3] + cb[3];
      float h4 = acc[mt][4] + cb[4], h5 = acc[mt][5] + cb[5];
      float h6 = acc[mt][6] + cb[6], h7 = acc[mt][7] + cb[7];
      float slo = h0 * h0 + h1 * h1 + h2 * h2 + h3 * h3;
      float shi = h4 * h4 + h5 * h5 + h6 * h6 + h7 * h7;
      yout[((size_t)b * Cdim + cc) * Mdim + m]     = sqrtf(slo) + pb0;
      yout[((size_t)b * Cdim + cc + 1) * Mdim + m] = sqrtf(shi) + pb1;
    }
  }
}

__global__ void copyx_k(const float* __restrict__ hist, float* __restrict__ xout) {
  long i = (long)blockIdx.x * 256 + threadIdx.x;
  if (i >= (long)Bdim * MC) return;
  long b = i / MC, r = i % MC;
  xout[i] = hist[(b * HISTN + (HISTN - 1)) * (long)MC + r];
}

// ---------------------------------------------------------------------------
extern "C" void kernel_launch(void* const* d_in, const int* in_sizes, int n_in,
                              void* d_out, int out_size, void* d_ws, size_t ws_size,
                              hipStream_t stream) {
  (void)in_sizes; (void)n_in; (void)out_size; (void)ws_size;
  const float* x_in  = (const float*)d_in[0];
  const float* y_in  = (const float*)d_in[1];
  const float* adj   = (const float*)d_in[2];
  const float* omega = (const float*)d_in[3];
  const float* syncp = (const float*)d_in[4];
  const float* gamma = (const float*)d_in[5];
  const float* beta  = (const float*)d_in[6];
  const float* convw = (const float*)d_in[7];
  const float* convb = (const float*)d_in[8];
  const float* phib  = (const float*)d_in[9];

  float* out   = (float*)d_out;
  float* x_out = out;                       // [B,M,C]
  float* y_out = out + (size_t)Bdim * MC;   // [B,C,M]
  float* hist  = y_out + (size_t)Bdim * MC; // [B,24,M,C]

  float* Pws = (float*)d_ws;                // 128*116 padded P
  float* yt  = Pws + 128 * Mdim;            // [B,M,C] groupnormed-transposed y

  const size_t STEP_LDS = (size_t)(128 * Mdim + 2 * Mdim * 128 + 256 + 128 + 64) * 4;
  const size_t CONV_LDS = (size_t)(128 * 514) * 4;  // 263168 B

  prep_p_k<<<(128 * Mdim + 255) / 256, 256, 0, stream>>>(syncp, adj, Pws);

  for (int l = 0; l < 3; ++l) {
    const float* ysrc = (l == 0) ? y_in : y_out;
    groupnorm_k<<<Bdim * 128, 32, 0, stream>>>(ysrc, gamma, beta, yt);
    for (int t = 0; t < 8; ++t) {
      int s = l * 8 + t;
      const float* xsb; long xsbs;
      if (s == 0) { xsb = x_in; xsbs = MC; }
      else        { xsb = hist + (size_t)(s - 1) * MC; xsbs = (long)HISTN * MC; }
      float* xdb = hist + (size_t)s * MC;
      step_k<<<dim3(Bdim, Cdim / 128), 256, STEP_LDS, stream>>>(
          Pws, yt, omega, xsb, xsbs, xdb, (long)HISTN * MC, (t == 0) ? 1 : 0);
    }
    conv_k<<<dim3(Bdim, 2048 / 128), 256, CONV_LDS, stream>>>(
        hist + (size_t)(l * 8 + 7) * MC, (long)HISTN * MC, convw, convb, phib, y_out);
  }
  copyx_k<<<((long)Bdim * MC + 255) / 256, 256, 0, stream>>>(hist, x_out);
}